// DMF_41540923687493
// MI455X (gfx1250) — compile-verified
//
#include <hip/hip_runtime.h>
#include <math.h>

#define N      1600
#define MAXD   500
#define BATCH  4
#define TSTEPS 500
#define DT     0.05f

typedef __attribute__((ext_vector_type(2))) float v2f;
typedef __attribute__((ext_vector_type(8))) float v8f;

// ---------------------------------------------------------------------------
// K0: Cw[j][d] = (#{k : delays[j][k]==d}) / N   (delays constant -> built once
// per launch, deterministically). One block per row, LDS histogram.
// ---------------------------------------------------------------------------
__global__ __launch_bounds__(256) void k_build_w(const int* __restrict__ delays,
                                                 float* __restrict__ Cw) {
  __shared__ unsigned cnt[MAXD];
  const int j = blockIdx.x;
  for (int d = threadIdx.x; d < MAXD; d += 256) cnt[d] = 0u;
  __syncthreads();
  const int* row = delays + (size_t)j * N;
  for (int k = threadIdx.x; k < N; k += 256) atomicAdd(&cnt[row[k]], 1u);
  __syncthreads();
  float* wrow = Cw + (size_t)j * MAXD;
  for (int d = threadIdx.x; d < MAXD; d += 256)
    wrow[d] = (float)cnt[d] * (1.0f / (float)N);
}

// ---------------------------------------------------------------------------
// K_init: circular history buffer <- hE input (head=0 => identity layout),
// live state <- x0. (Inputs are never mutated.)
// ---------------------------------------------------------------------------
__global__ void k_init(const float* __restrict__ hE, const float* __restrict__ x0,
                       float* __restrict__ buf, float* __restrict__ xst) {
  int i = blockIdx.x * blockDim.x + threadIdx.x;
  if (i < N * MAXD) buf[i] = hE[i];
  if (i < N * 2)    xst[i] = x0[i];
}

// K_copyx: snapshot current state into X[b*TSTEPS + 0]
__global__ void k_copyx(const float* __restrict__ xst, float* __restrict__ dst) {
  int i = blockIdx.x * blockDim.x + threadIdx.x;
  if (i < N * 2) dst[i] = xst[i];
}

// ---------------------------------------------------------------------------
// K1: delayed_E[j] = sum_d Cw[j][d] * buf[j][(head+d) % MAXD]
// One wave per row; coalesced reads; wave32 shuffle reduction.
// ---------------------------------------------------------------------------
__global__ __launch_bounds__(256) void k_delayed(const float* __restrict__ Cw,
                                                 const float* __restrict__ buf,
                                                 float* __restrict__ dEv,
                                                 int head) {
  const int wave = threadIdx.x >> 5, lane = threadIdx.x & 31;
  const int j = blockIdx.x * 8 + wave;
  const float* w = Cw  + (size_t)j * MAXD;
  const float* b = buf + (size_t)j * MAXD;
  float acc = 0.0f;
  for (int d = lane; d < MAXD; d += 32) {
    int idx = d + head; if (idx >= MAXD) idx -= MAXD;
    acc = fmaf(w[d], b[idx], acc);
  }
  for (int off = 16; off > 0; off >>= 1) acc += __shfl_xor(acc, off, 32);
  if (lane == 0) dEv[j] = acc;
}

// ---------------------------------------------------------------------------
// K2: Lv = L @ delayed_E via v_wmma_f32_16x16x4_f32, then pointwise DMF update.
// One 16-row tile per block; K split across 8 waves (50 WMMAs each, 2 accums).
// B operand: delayed_E segment replicated across all 16 columns (LDS broadcast
// reads), so every D column holds the same dot product.
//   A 16x4 f32 layout: lanes 0-15 M=0..15 {K0,K1}; lanes 16-31 {K2,K3}
//   B 4x16 f32 layout (mirrored): VGPR0 rows K0/K2, VGPR1 rows K1/K3
//   D: vgpr v -> row v (lanes 0-15) / row v+8 (lanes 16-31)
// ---------------------------------------------------------------------------
__global__ __launch_bounds__(256) void k_step(const float* __restrict__ L,
                                              const float* __restrict__ dEv,
                                              const float* __restrict__ theta,
                                              const float* __restrict__ noise_raw,
                                              float* __restrict__ xst,
                                              float* __restrict__ buf,
                                              float* __restrict__ Xout,
                                              int b, int newHead) {
  __shared__ float sv[N];        // delayed_E vector
  __shared__ float s_lv[8 * 16]; // per-wave partial row sums
  const int tid  = threadIdx.x;
  const int wave = tid >> 5, lane = tid & 31;
  const int i0 = blockIdx.x * 16;

  for (int i = tid; i < N; i += 256) sv[i] = dEv[i];
  __syncthreads();

  const int aoff = (lane >> 4) << 1;           // 0 for lanes 0-15, 2 for 16-31
  const int row  = i0 + (lane & 15);
  const float* Lr = L + (size_t)row * N;
  v8f c0 = {}; v8f c1 = {};
  const int kBase = wave * 200;                // 200 K per wave, blocks of 4
  #pragma unroll 5
  for (int kk = 0; kk < 200; kk += 8) {
    const int k0 = kBase + kk;
    v2f a0 = *(const v2f*)(Lr + k0 + aoff);
    v2f b0 = *(const v2f*)(&sv[k0 + aoff]);
    v2f a1 = *(const v2f*)(Lr + k0 + 4 + aoff);
    v2f b1 = *(const v2f*)(&sv[k0 + 4 + aoff]);
    c0 = __builtin_amdgcn_wmma_f32_16x16x4_f32(false, a0, false, b0,
                                               (short)0, c0, false, false);
    c1 = __builtin_amdgcn_wmma_f32_16x16x4_f32(false, a1, false, b1,
                                               (short)0, c1, false, false);
  }
  c0 = c0 + c1;

  if (lane == 0) {
    #pragma unroll
    for (int v = 0; v < 8; ++v) s_lv[wave * 16 + v] = c0[v];
  } else if (lane == 16) {
    #pragma unroll
    for (int v = 0; v < 8; ++v) s_lv[wave * 16 + 8 + v] = c0[v];
  }
  __syncthreads();

  if (tid < 16) {
    const int i = i0 + tid;
    float Lv = 0.0f;
    #pragma unroll
    for (int w = 0; w < 8; ++w) Lv += s_lv[w * 16 + tid];

    const float std_in = theta[0], W_E = theta[1],  tau_E = theta[2];
    const float gamma_E = theta[3], W_I = theta[4], tau_I = theta[5];
    const float I_0 = theta[6],  g = theta[7],   g_EE = theta[8];
    const float g_IE = theta[9], g_EI = theta[10];
    const float aE = theta[11], bE = theta[12], dEc = theta[13];
    const float aI = theta[14], bI = theta[15], dIc = theta[16];

    const float E = xst[i * 2 + 0];
    const float I = xst[i * 2 + 1];
    const float I_E = W_E * I_0 + g_EE * E + g * Lv - g_IE * I;
    const float I_I = W_I * I_0 + g_EI * E - I;

    const float uE = aE * I_E - bE;
    const float RE = (1e-5f + fabsf(uE)) /
                     (1e-5f * dEc + fabsf(1.0f - expf(-dEc * uE)) + 1e-8f);
    const float uI = aI * I_I - bI;
    const float RI = (1e-5f + fabsf(uI)) /
                     (1e-5f * dIc + fabsf(1.0f - expf(-dIc * uI)) + 1e-8f);

    const float dE_ = -E / tau_E + (1.0f - E) * gamma_E * RE;
    const float dI_ = -I / tau_I + RI;
    const float sdt = std_in * sqrtf(DT);
    const float En = E + DT * dE_ + noise_raw[((size_t)b * N + i) * 2 + 0] * sdt;
    const float In = I + DT * dI_ + noise_raw[((size_t)b * N + i) * 2 + 1] * sdt;

    xst[i * 2 + 0] = En;  xst[i * 2 + 1] = In;
    Xout[i * 2 + 0] = En; Xout[i * 2 + 1] = In;
    buf[(size_t)i * MAXD + newHead] = En;     // history push (circular)
  }
}

// ---------------------------------------------------------------------------
// K_final: rotate circular buffer back to reference layout:
// hE_f[j][d] = buf[j][(head+d) % MAXD], in place (block-per-row, barrier).
// ---------------------------------------------------------------------------
__global__ __launch_bounds__(512) void k_unrot(float* __restrict__ buf, int head) {
  const int j = blockIdx.x, tid = threadIdx.x;
  float v = 0.0f;
  if (tid < MAXD) {
    int idx = tid + head; if (idx >= MAXD) idx -= MAXD;
    v = buf[(size_t)j * MAXD + idx];
  }
  __syncthreads();
  if (tid < MAXD) buf[(size_t)j * MAXD + tid] = v;
}

// ---------------------------------------------------------------------------
extern "C" void kernel_launch(void* const* d_in, const int* in_sizes, int n_in,
                              void* d_out, int out_size, void* d_ws, size_t ws_size,
                              hipStream_t stream) {
  const float* x0     = (const float*)d_in[0];
  const float* hE     = (const float*)d_in[1];
  const float* L      = (const float*)d_in[2];
  const float* theta  = (const float*)d_in[3];
  const float* noise  = (const float*)d_in[4];
  const int*   delays = (const int*)d_in[5];

  float* out = (float*)d_out;
  // d_out layout: X_all (B*T*N*2) | x_f (N*2) | hE_f (N*MAXD)
  float* xst = out + (size_t)BATCH * TSTEPS * N * 2; // live state -> becomes x_f
  float* buf = xst + (size_t)N * 2;                  // circular hist -> hE_f
  // workspace: Cw (N*MAXD f32 = 3.2MB) | delayed_E (N f32)
  float* Cw  = (float*)d_ws;
  float* dEv = Cw + (size_t)N * MAXD;

  k_build_w<<<N, 256, 0, stream>>>(delays, Cw);
  k_init<<<(N * MAXD + 255) / 256, 256, 0, stream>>>(hE, x0, buf, xst);

  int head = 0; // conceptual delay d  <->  slot (head+d) % MAXD
  for (int b = 0; b < BATCH; ++b) {
    k_copyx<<<(N * 2 + 255) / 256, 256, 0, stream>>>(
        xst, out + (size_t)b * TSTEPS * N * 2);
    for (int t = 1; t < TSTEPS; ++t) {
      int nh = head - 1; if (nh < 0) nh += MAXD;
      k_delayed<<<N / 8, 256, 0, stream>>>(Cw, buf, dEv, head);
      k_step<<<N / 16, 256, 0, stream>>>(
          L, dEv, theta, noise, xst, buf,
          out + ((size_t)b * TSTEPS + t) * N * 2, b, nh);
      head = nh;
    }
  }
  k_unrot<<<N, 512, 0, stream>>>(buf, head);
}